// UnivariateFlowMixture_9783935500639
// MI455X (gfx1250) — compile-verified
//
#include <hip/hip_runtime.h>
#include <stdint.h>

// ---------------------------------------------------------------------------
// Rational-quadratic spline flow: N samples, L=8 layers, C=8 channels, B=32.
// prep: build ready-to-eval tables (edges, cum-heights, derivs, per-bin
//       {invw, -xk*invw, dk+dk1-2*delta, delta} records, fused affine) in d_ws.
// main: async-stage tables to LDS (CDNA5 global_load_async_to_lds_b128 /
//       s_wait_asynccnt), then 1 sample/thread x 8 channels x 8 layers with a
//       5-step LDS binary search and FMA-heavy rational-quadratic eval.
// ---------------------------------------------------------------------------

namespace {
constexpr int   NL = 8, NC = 8, NB = 32;
constexpr float TB = 3.0f;                        // TAIL_BOUND
// per-(layer,channel) table layout (floats):
constexpr int   E_OFF  = 0;                       // 33 search edges (last +1e-6)
constexpr int   CH_OFF = 33;                      // 33 cumulative heights
constexpr int   D_OFF  = 66;                      // 33 derivatives
constexpr int   RA_OFF = 100;                     // 32 x {invw, nxk, c2, delta}, 16B aligned
constexpr int   STRIDE = 228;                     // 100 + 128 (lc*228*4 = 912B, 16B-aligned)
constexpr int   TBL_FLOATS = NL * NC * STRIDE;    // 14592 floats = 58368 B LDS
constexpr int   AFF_S  = TBL_FLOATS;              // 72: exp(-log_scale)
constexpr int   AFF_NB = TBL_FLOATS + 72;         // 72: -bias * exp(-log_scale)
constexpr int   AFF_LS = TBL_FLOATS + 144;        // 8:  per-channel sum of log_scale
}

__global__ void rq_flow_prep(const float* __restrict__ w, const float* __restrict__ h,
                             const float* __restrict__ s, const float* __restrict__ bias,
                             const float* __restrict__ lsc, float* __restrict__ ws) {
  const int t = threadIdx.x;
  if (t < (NL + 1) * NC) {                        // 72 fused-affine entries
    const float si = __expf(-lsc[t]);
    ws[AFF_S + t]  = si;
    ws[AFF_NB + t] = -bias[t] * si;
  }
  if (t < NC) {                                   // per-channel total log_scale
    float acc = 0.f;
    for (int l = 0; l <= NL; ++l) acc += lsc[l * NC + t];
    ws[AFF_LS + t] = acc;
  }
  if (t >= NL * NC) return;                       // one thread per (layer,channel)

  const float* uw = w + t * NB;
  const float* uh = h + t * NB;
  const float* us = s + t * (NB - 1);
  float* T = ws + t * STRIDE;
  const float sc = 1.0f - 1e-3f * NB;

  float eL[NB + 1], chL[NB + 1], dL[NB + 1];

  // ---- widths -> x-edges ----
  {
    float m = -3.4e38f;
    for (int j = 0; j < NB; ++j) m = fmaxf(m, uw[j]);
    float sum = 0.f;
    for (int j = 0; j < NB; ++j) sum += __expf(uw[j] - m);
    const float inv = 1.f / sum;
    float cum = 0.f;
    eL[0] = -TB;
    for (int j = 0; j < NB; ++j) {
      cum += 1e-3f + sc * __expf(uw[j] - m) * inv;
      eL[j + 1] = (j == NB - 1) ? TB : (2.f * TB * cum - TB);
    }
  }
  // ---- heights -> cumulative heights ----
  {
    float m = -3.4e38f;
    for (int j = 0; j < NB; ++j) m = fmaxf(m, uh[j]);
    float sum = 0.f;
    for (int j = 0; j < NB; ++j) sum += __expf(uh[j] - m);
    const float inv = 1.f / sum;
    float cum = 0.f;
    chL[0] = -TB;
    for (int j = 0; j < NB; ++j) {
      cum += 1e-3f + sc * __expf(uh[j] - m) * inv;
      chL[j + 1] = (j == NB - 1) ? TB : (2.f * TB * cum - TB);
    }
  }
  // ---- derivatives: MIN_DERIV + softplus(logit) ----
  {
    const float cst = logf(expm1f(1.0f - 1e-3f));
    const float dedge = 1e-3f + ((cst > 20.f) ? cst : log1pf(__expf(cst)));
    dL[0] = dedge;
    dL[NB] = dedge;
    for (int j = 1; j < NB; ++j) {
      const float u = us[j - 1];
      dL[j] = 1e-3f + ((u > 20.f) ? u : log1pf(__expf(u)));
    }
  }
  // ---- emit tables ----
  for (int j = 0; j < NB; ++j) T[E_OFF + j] = eL[j];
  T[E_OFF + NB] = TB + 1e-6f;                     // eps-shifted last search edge
  for (int j = 0; j <= NB; ++j) T[CH_OFF + j] = chL[j];
  for (int j = 0; j <= NB; ++j) T[D_OFF + j]  = dL[j];
  for (int j = 0; j < NB; ++j) {
    const float invw = 1.f / (eL[j + 1] - eL[j]);
    const float hk   = chL[j + 1] - chL[j];
    const float dl   = hk * invw;                 // delta = h/w
    T[RA_OFF + 4 * j + 0] = invw;
    T[RA_OFF + 4 * j + 1] = -eL[j] * invw;        // nxk
    T[RA_OFF + 4 * j + 2] = dL[j] + dL[j + 1] - 2.f * dl;  // c2
    T[RA_OFF + 4 * j + 3] = dl;
  }
}

__global__ void __launch_bounds__(256) rq_flow_main(
    const float* __restrict__ x, const float* __restrict__ ws,
    float* __restrict__ out, int n) {
  __shared__ float tbl[TBL_FLOATS];

  // ---- async global -> LDS staging (ASYNCcnt-tracked CDNA5 path) ----
  {
    const uint32_t lbase = (uint32_t)(uintptr_t)(&tbl[0]);
    for (int j = (int)threadIdx.x; j < TBL_FLOATS / 4; j += 256) {
      uint32_t loff = lbase + (uint32_t)j * 16u;
      uint32_t goff = (uint32_t)j * 16u;
      asm volatile("global_load_async_to_lds_b128 %0, %1, %2"
                   :: "v"(loff), "v"(goff), "s"(ws) : "memory");
    }
    asm volatile("s_wait_asynccnt 0" ::: "memory");
  }
  __syncthreads();

  const int i = (int)(blockIdx.x * blockDim.x + threadIdx.x);
  if (i >= n) return;

  float z[NC], lj[NC];
  {
    const float xv = x[i];
#pragma unroll
    for (int c = 0; c < NC; ++c) { z[c] = xv; lj[c] = 0.f; }
  }

#pragma unroll 1
  for (int l = 0; l < NL; ++l) {
#pragma unroll
    for (int c = 0; c < NC; ++c) {
      const int lc = l * NC + c;
      // fused affine (wave-uniform -> scalar loads): zz = (z-b)*exp(-ls)
      const float as = ws[AFF_S + lc];
      const float nb = ws[AFF_NB + lc];
      const float zz = __builtin_fmaf(z[c], as, nb);

      const float xc = fminf(fmaxf(zz, -TB), TB);
      const bool inside = (zz == xc);             // NaN -> false (pass-through)

      const float* T = &tbl[lc * STRIDE];
      // 5-step binary search over 33 edges (idx bits are disjoint -> OR)
      int idx = (xc >= T[16]) ? 16 : 0;
      idx |= (xc >= T[idx + 8]) ? 8 : 0;
      idx |= (xc >= T[idx + 4]) ? 4 : 0;
      idx |= (xc >= T[idx + 2]) ? 2 : 0;
      idx |= (xc >= T[idx + 1]) ? 1 : 0;

      const float4 ra  = ((const float4*)(T + RA_OFF))[idx];  // ds_load_b128
      const float  invw = ra.x, nxk = ra.y, c2 = ra.z, dl = ra.w;
      const float  yk   = T[CH_OFF + idx];
      const float  yk1  = T[CH_OFF + idx + 1];    // pairs with yk (2addr)
      const float  dk   = T[D_OFF + idx];
      const float  dk1  = T[D_OFF + idx + 1];     // pairs with dk (2addr)

      const float hk   = yk1 - yk;
      const float th   = __builtin_fmaf(xc, invw, nxk);
      const float omt  = 1.f - th;
      const float t1m  = th * omt;
      const float th2  = th * th;
      const float omt2 = omt * omt;

      const float num  = hk * __builtin_fmaf(dk, t1m, dl * th2);
      const float den  = __builtin_fmaf(c2, t1m, dl);
      const float rden = __builtin_amdgcn_rcpf(den);
      const float y    = __builtin_fmaf(num, rden, yk);

      const float tdl  = dl + dl;
      const float acc  = __builtin_fmaf(dk, omt2,
                         __builtin_fmaf(tdl, t1m, dk1 * th2));
      const float dnum = (dl * dl) * acc;
      const float ld   = __logf(dnum * (rden * rden)); // log(dnum)-2*log(den)

      z[c]   = inside ? y : zz;
      lj[c] += inside ? ld : 0.f;
    }
  }

  // final affine layer (l = NL) + fold total log_scale sum
#pragma unroll
  for (int c = 0; c < NC; ++c) {
    const int lc = NL * NC + c;
    z[c]  = __builtin_fmaf(z[c], ws[AFF_S + lc], ws[AFF_NB + lc]);
    lj[c] -= ws[AFF_LS + c];
  }

  float4* oz = reinterpret_cast<float4*>(out) + (size_t)i * 2;
  float4* ol = reinterpret_cast<float4*>(out + (size_t)n * NC) + (size_t)i * 2;
  oz[0] = make_float4(z[0], z[1], z[2], z[3]);
  oz[1] = make_float4(z[4], z[5], z[6], z[7]);
  ol[0] = make_float4(lj[0], lj[1], lj[2], lj[3]);
  ol[1] = make_float4(lj[4], lj[5], lj[6], lj[7]);
}

extern "C" void kernel_launch(void* const* d_in, const int* in_sizes, int n_in,
                              void* d_out, int out_size, void* d_ws, size_t ws_size,
                              hipStream_t stream) {
  (void)n_in; (void)out_size; (void)ws_size;
  const float* x   = (const float*)d_in[0];
  const float* w   = (const float*)d_in[1];
  const float* h   = (const float*)d_in[2];
  const float* s   = (const float*)d_in[3];
  const float* b   = (const float*)d_in[4];
  const float* lsc = (const float*)d_in[5];
  float* ws  = (float*)d_ws;
  float* out = (float*)d_out;
  const int n = in_sizes[0];

  rq_flow_prep<<<1, 128, 0, stream>>>(w, h, s, b, lsc, ws);
  const int blocks = (n + 255) / 256;
  rq_flow_main<<<blocks, 256, 0, stream>>>(x, ws, out, n);
}